// TULayer_2396591751780
// MI455X (gfx1250) — compile-verified
//
#include <hip/hip_runtime.h>
#include <hip/hip_bf16.h>

typedef float v2f __attribute__((ext_vector_type(2)));
typedef float v4f __attribute__((ext_vector_type(4)));
typedef float v8f __attribute__((ext_vector_type(8)));

#define B_    2
#define M_    4096
#define N_    16384
#define CIN_  256
#define COUT_ 256

// ---------------------------------------------------------------------------
// GEMM 1: p1t[b][m][o] = sum_c w1[o][c] * pts1[b][c][m] + b1[o]
// One wave per 16(o) x 16(m) tile, V_WMMA_F32_16X16X4_F32 over K=CIN in steps of 4.
// A-layout (16x4 f32): lane(0..15)=row, {lane>=16, vgpr} select K 0..3.
// B-layout (4x16 f32): lane%16=col,   {lane>=16, vgpr} select K 0..3.
// D-layout: vgpr v, half h -> row o0+8h+v ; col = lane%16.
// ---------------------------------------------------------------------------
__global__ __launch_bounds__(256) void gemm_p1_kernel(
    const float* __restrict__ w1, const float* __restrict__ b1,
    const float* __restrict__ pts1,   // [B, CIN, M]
    float* __restrict__ p1t)          // [B, M, COUT]
{
  const int lane = threadIdx.x & 31;
  const int wave = threadIdx.x >> 5;
  const int tile = blockIdx.x * 8 + wave;             // 8192 tiles total
  const int tilesPerB = (COUT_ / 16) * (M_ / 16);     // 16 * 256
  const int bb = tile / tilesPerB;
  const int t  = tile % tilesPerB;
  const int o0 = (t / (M_ / 16)) * 16;
  const int m0 = (t % (M_ / 16)) * 16;

  const int half = lane >> 4;        // 0 or 1  -> K-half
  const int lid  = lane & 15;
  const int arow = o0 + lid;         // output channel (A row)
  const int bcol = m0 + lid;         // point index   (B col)

  v8f acc = {};
  for (int c = 0; c < CIN_; c += 4) {
    const int k0 = c + 2 * half;
    v2f a, bm;
    a.x  = w1[arow * CIN_ + k0];
    a.y  = w1[arow * CIN_ + k0 + 1];
    bm.x = pts1[((size_t)bb * CIN_ + k0)     * M_ + bcol];
    bm.y = pts1[((size_t)bb * CIN_ + k0 + 1) * M_ + bcol];
    acc = __builtin_amdgcn_wmma_f32_16x16x4_f32(
        /*neg_a=*/false, a, /*neg_b=*/false, bm,
        /*c_mod=*/(short)0, acc, /*reuse_a=*/false, /*reuse_b=*/false);
  }

  const int obase = o0 + 8 * half;
  float out8[8];
#pragma unroll
  for (int v = 0; v < 8; ++v) out8[v] = acc[v] + b1[obase + v];

  // 8 contiguous channels per lane -> two 16B stores
  float* dst = p1t + (((size_t)bb * M_ + (m0 + lid)) * COUT_ + obase);
  *(v4f*)(dst)     = *(const v4f*)(&out8[0]);
  *(v4f*)(dst + 4) = *(const v4f*)(&out8[4]);
}

// ---------------------------------------------------------------------------
// GEMM 2: out2[b][o][n] = sum_c w2[o][c] * pts2[b][c][n] + b2[o]
// Written directly into the output region (interp kernel adds on top later).
// ---------------------------------------------------------------------------
__global__ __launch_bounds__(256) void gemm_p2_kernel(
    const float* __restrict__ w2, const float* __restrict__ b2,
    const float* __restrict__ pts2,   // [B, COUT(in c-role: CIN==COUT), N]
    float* __restrict__ out2)         // [B, COUT, N]
{
  const int lane = threadIdx.x & 31;
  const int wave = threadIdx.x >> 5;
  const int tile = blockIdx.x * 8 + wave;             // 32768 tiles total
  const int tilesPerB = (COUT_ / 16) * (N_ / 16);     // 16 * 1024
  const int bb = tile / tilesPerB;
  const int t  = tile % tilesPerB;
  const int o0 = (t / (N_ / 16)) * 16;
  const int n0 = (t % (N_ / 16)) * 16;

  const int half = lane >> 4;
  const int lid  = lane & 15;
  const int arow = o0 + lid;
  const int bcol = n0 + lid;

  v8f acc = {};
  for (int c = 0; c < COUT_; c += 4) {
    const int k0 = c + 2 * half;
    v2f a, bm;
    a.x  = w2[arow * COUT_ + k0];
    a.y  = w2[arow * COUT_ + k0 + 1];
    bm.x = pts2[((size_t)bb * COUT_ + k0)     * N_ + bcol];
    bm.y = pts2[((size_t)bb * COUT_ + k0 + 1) * N_ + bcol];
    acc = __builtin_amdgcn_wmma_f32_16x16x4_f32(
        false, a, false, bm, (short)0, acc, false, false);
  }

  const int obase = o0 + 8 * half;
#pragma unroll
  for (int v = 0; v < 8; ++v) {
    const int o = obase + v;
    out2[((size_t)bb * COUT_ + o) * N_ + (n0 + lid)] = acc[v] + b2[o];
  }
}

// ---------------------------------------------------------------------------
// kNN: for each query point (b,n) find the 3 nearest of the M source points.
// xyz1[b] (48 KB) staged in LDS; each thread scans all M for one n.
// Same-iteration LDS reads are a broadcast (conflict-free).
// ---------------------------------------------------------------------------
__global__ __launch_bounds__(256) void knn_kernel(
    const float* __restrict__ xyz1,   // [B, 3, M]
    const float* __restrict__ xyz2,   // [B, 3, N]
    int*   __restrict__ knn_idx,      // [B*N, 3]
    float* __restrict__ knn_w)        // [B*N, 3]
{
  __shared__ float sx[M_];
  __shared__ float sy[M_];
  __shared__ float sz[M_];

  const int blocksPerB = N_ / 256;    // 64
  const int b = blockIdx.x / blocksPerB;
  const int n = (blockIdx.x % blocksPerB) * 256 + threadIdx.x;

  const float* base = xyz1 + (size_t)b * 3 * M_;
  for (int m = threadIdx.x; m < M_; m += 256) {
    sx[m] = base[m];
    sy[m] = base[M_ + m];
    sz[m] = base[2 * M_ + m];
  }
  __syncthreads();

  const float qx = xyz2[((size_t)b * 3 + 0) * N_ + n];
  const float qy = xyz2[((size_t)b * 3 + 1) * N_ + n];
  const float qz = xyz2[((size_t)b * 3 + 2) * N_ + n];

  float d0 = 3.4e38f, d1 = 3.4e38f, d2 = 3.4e38f;
  int   i0 = 0, i1 = 0, i2 = 0;

#pragma unroll 4
  for (int m = 0; m < M_; ++m) {
    const float dx = sx[m] - qx;
    const float dy = sy[m] - qy;
    const float dz = sz[m] - qz;
    const float d = dx * dx + dy * dy + dz * dz;
    if (d < d0)      { d2 = d1; i2 = i1; d1 = d0; i1 = i0; d0 = d; i0 = m; }
    else if (d < d1) { d2 = d1; i2 = i1; d1 = d;  i1 = m; }
    else if (d < d2) { d2 = d;  i2 = m; }
  }

  const float r0 = 1.0f / (d0 + 1e-8f);
  const float r1 = 1.0f / (d1 + 1e-8f);
  const float r2 = 1.0f / (d2 + 1e-8f);
  const float inv = 1.0f / (r0 + r1 + r2);

  const size_t g = (size_t)b * N_ + n;
  knn_idx[g * 3 + 0] = i0;  knn_w[g * 3 + 0] = r0 * inv;
  knn_idx[g * 3 + 1] = i1;  knn_w[g * 3 + 1] = r1 * inv;
  knn_idx[g * 3 + 2] = i2;  knn_w[g * 3 + 2] = r2 * inv;
}

// ---------------------------------------------------------------------------
// Interpolate: out2[b][c][n] += sum_k w_k * p1t[b][idx_k][c]
// One block per (b,n); thread c handles channel c (gather coalesced along c).
// ---------------------------------------------------------------------------
__global__ __launch_bounds__(256) void interp_kernel(
    const float* __restrict__ p1t,    // [B, M, COUT]
    const int*   __restrict__ knn_idx,
    const float* __restrict__ knn_w,
    float* __restrict__ out2)         // [B, COUT, N]
{
  const int gid = blockIdx.x;         // b*N + n
  const int b = gid / N_;
  const int n = gid % N_;
  const int c = threadIdx.x;

  const int   j0 = knn_idx[(size_t)gid * 3 + 0];
  const int   j1 = knn_idx[(size_t)gid * 3 + 1];
  const int   j2 = knn_idx[(size_t)gid * 3 + 2];
  const float u0 = knn_w[(size_t)gid * 3 + 0];
  const float u1 = knn_w[(size_t)gid * 3 + 1];
  const float u2 = knn_w[(size_t)gid * 3 + 2];

  const float* pb = p1t + (size_t)b * M_ * COUT_;
  const float acc = u0 * pb[(size_t)j0 * COUT_ + c]
                  + u1 * pb[(size_t)j1 * COUT_ + c]
                  + u2 * pb[(size_t)j2 * COUT_ + c];
  out2[((size_t)b * COUT_ + c) * N_ + n] += acc;
}

// ---------------------------------------------------------------------------
// Passthrough: first tuple output is xyz_2 itself.
// ---------------------------------------------------------------------------
__global__ __launch_bounds__(256) void copy_xyz_kernel(
    const float* __restrict__ xyz2, float* __restrict__ out)
{
  const int i = blockIdx.x * 256 + threadIdx.x;
  out[i] = xyz2[i];
}

extern "C" void kernel_launch(void* const* d_in, const int* in_sizes, int n_in,
                              void* d_out, int out_size, void* d_ws, size_t ws_size,
                              hipStream_t stream) {
  const float* xyz1 = (const float*)d_in[0];   // [B,3,M]
  const float* xyz2 = (const float*)d_in[1];   // [B,3,N]
  const float* pts1 = (const float*)d_in[2];   // [B,CIN,M]
  const float* pts2 = (const float*)d_in[3];   // [B,COUT,N]
  const float* w1   = (const float*)d_in[4];   // [COUT,CIN]
  const float* b1   = (const float*)d_in[5];   // [COUT]
  const float* w2   = (const float*)d_in[6];   // [COUT,COUT]
  const float* b2   = (const float*)d_in[7];   // [COUT]

  float* out  = (float*)d_out;                 // [B*3*N] then [B,COUT,N]
  float* out2 = out + (size_t)B_ * 3 * N_;

  // Workspace layout
  float* p1t     = (float*)d_ws;                                  // B*M*COUT f32 (8 MB)
  int*   knn_idx = (int*)  ((char*)d_ws + (size_t)B_ * M_ * COUT_ * sizeof(float));
  float* knn_w   = (float*)((char*)knn_idx + (size_t)B_ * N_ * 3 * sizeof(int));

  // 1) p1 GEMM -> workspace (point-major for the gather)
  gemm_p1_kernel<<<(B_ * (COUT_ / 16) * (M_ / 16)) / 8, 256, 0, stream>>>(w1, b1, pts1, p1t);

  // 2) p2 GEMM -> straight into output region
  gemm_p2_kernel<<<(B_ * (COUT_ / 16) * (N_ / 16)) / 8, 256, 0, stream>>>(w2, b2, pts2, out2);

  // 3) fused distance + top-3 kNN + weights
  knn_kernel<<<B_ * (N_ / 256), 256, 0, stream>>>(xyz1, xyz2, knn_idx, knn_w);

  // 4) weighted gather-add on top of p2
  interp_kernel<<<B_ * N_, 256, 0, stream>>>(p1t, knn_idx, knn_w, out2);

  // 5) xyz_2 passthrough (first tuple output)
  copy_xyz_kernel<<<(B_ * 3 * N_) / 256, 256, 0, stream>>>(xyz2, out);
}